// CapsuleLayer_33698313404835
// MI455X (gfx1250) — compile-verified
//
#include <hip/hip_runtime.h>
#include <math.h>

// ---------------- problem constants (from reference) ----------------
#define BATCH 16
#define CI    512
#define CO    128
#define QS    64        // 8x8 pose votes per capsule pair
#define IC    4         // capsules-in staged per LDS chunk
#define NCHUNK (CI / IC)
#define LAMB  0.2f
#define HALF_PLUS 1.4189385332046727f   // 0.5 + 0.5*log(2*pi)
#define SQRT_2PI  2.5066282746310002f

typedef float  v2f __attribute__((ext_vector_type(2)));
typedef float  v8f __attribute__((ext_vector_type(8)));

// ------------------------------------------------------------------
// CDNA5 async global->LDS copies (ASYNCcnt-tracked). The generic pointer's
// low 32 bits are the LDS byte offset (aperture check truncates to 32b).
// ------------------------------------------------------------------
__device__ __forceinline__ void async_copy_b128(const float* lds_dst,
                                                const float* gsrc) {
    unsigned int l = (unsigned int)(unsigned long long)lds_dst;
    asm volatile("global_load_async_to_lds_b128 %0, %1, off"
                 :: "v"(l), "v"(gsrc) : "memory");
}
__device__ __forceinline__ void async_copy_b64(const float* lds_dst,
                                               const float* gsrc) {
    unsigned int l = (unsigned int)(unsigned long long)lds_dst;
    asm volatile("global_load_async_to_lds_b64 %0, %1, off"
                 :: "v"(l), "v"(gsrc) : "memory");
}
__device__ __forceinline__ void wait_async0() {
    asm volatile("s_wait_asynccnt 0" ::: "memory");
}

// Stage one chunk of IC capsules: transitions tile [IC][16 o][64] (16 KB) and
// pose tile [IC][2 b][64] (2 KB). 256 threads; per thread 4x b128 + 1x b64.
__device__ __forceinline__ void stage_chunk(const float* __restrict__ trans,
                                            const float* __restrict__ pose,
                                            float* ldsT, float* ldsP,
                                            int t, int bp, int obBase, int i0) {
    const int ii = t >> 6;               // capsule row within chunk
    const int tt = t & 63;
    const float* gRow = trans + ((long)(i0 + ii) * CO + obBase) * QS;
    float* lRow = ldsT + ii * (16 * QS);
#pragma unroll
    for (int rep = 0; rep < 4; ++rep) {
        const int off = (rep * 64 + tt) * 4;   // float index, 16B granules
        async_copy_b128(lRow + off, gRow + off);
    }
    const int e = t * 2;                 // 2 floats per thread (b64)
    const int pii  = e >> 7;
    const int bsel = (e >> 6) & 1;
    const int w    = e & 63;
    const float* gp = pose + ((long)(bp * 2 + bsel) * CI + (i0 + pii)) * QS + w;
    async_copy_b64(ldsP + e, gp);
}

// ------------------------------------------------------------------
// WMMA tile plumbing: V[b,i,o,q,s] = sum_p pose[b,i,p,q] * T[i,o,p,s]
// A (16x4 f32): lane&15 = M = b_sel*8+q; VGPR0 K=(lane<16?0:2), VGPR1 K+1
// B (4x16 f32): lane&15 = N = o_sel*8+s; VGPR0 K=(lane<16?0:1), VGPR1 K+2
// C/D: VGPR j -> q=j; lane>>4 -> b_sel; (lane&15)>>3 -> o_sel; lane&7 -> s
// K = p = 8 as two chained 16x16x4 ops; 100% useful MACs.
// ------------------------------------------------------------------
struct LaneMap {
    int a00, a01, a10, a11;   // LDS offsets into pose chunk row (128 floats)
    int b00, b01, b10, b11;   // LDS offsets into trans chunk row (1024 floats)
    int bC, oC, sC;           // C/D-tile lane ownership (absolute b, o)
};

__device__ __forceinline__ LaneMap make_lane_map(int lane, int wave,
                                                 int bp, int obBase) {
    LaneMap L;
    const int mA = lane & 15;
    const int qa = mA & 7;
    const int aB = (mA >> 3) * QS;            // b_sel block in pose row
    const int kA = (lane >> 4) * 2;
    L.a00 = aB + (kA + 0) * 8 + qa;
    L.a01 = aB + (kA + 1) * 8 + qa;
    L.a10 = aB + (kA + 4) * 8 + qa;
    L.a11 = aB + (kA + 5) * 8 + qa;
    const int nB = lane & 15;
    const int sB = nB & 7;
    const int oRel = wave * 2 + (nB >> 3);    // o row within 16-o tile
    const int pb = lane >> 4;
    const int tB = oRel * QS + sB;
    L.b00 = tB + (pb + 0) * 8;
    L.b01 = tB + (pb + 2) * 8;
    L.b10 = tB + (pb + 4) * 8;
    L.b11 = tB + (pb + 6) * 8;
    L.bC = bp * 2 + (lane >> 4);
    L.oC = obBase + wave * 2 + ((lane & 15) >> 3);
    L.sC = lane & 7;
    return L;
}

__device__ __forceinline__ v8f vote_tile(const float* lp, const float* lt,
                                         const LaneMap& L) {
    v2f A0 = { lp[L.a00], lp[L.a01] };
    v2f A1 = { lp[L.a10], lp[L.a11] };
    v2f B0 = { lt[L.b00], lt[L.b01] };
    v2f B1 = { lt[L.b10], lt[L.b11] };
    v8f v = {};
    v = __builtin_amdgcn_wmma_f32_16x16x4_f32(false, A0, false, B0, (short)0, v, false, false);
    v = __builtin_amdgcn_wmma_f32_16x16x4_f32(false, A1, false, B1, (short)0, v, false, false);
    return v;
}

// ---------------- kernel: R <- 1/CO ----------------
__global__ void initR_kernel(float* __restrict__ R) {
    R[blockIdx.x * 256 + threadIdx.x] = 1.0f / (float)CO;
}

// ---------------- kernel: denom[b,o] = sum_i act[b,i]*R[b,i,o] ----------------
__global__ __launch_bounds__(CO) void denom_kernel(const float* __restrict__ act,
                                                   const float* __restrict__ R,
                                                   float* __restrict__ denom) {
    const int b = blockIdx.x, o = threadIdx.x;
    const float* Rr = R + (long)b * CI * CO + o;
    const float* ar = act + b * CI;
    float s = 0.f;
    for (int i = 0; i < CI; ++i) s += ar[i] * Rr[(long)i * CO];
    denom[b * CO + o] = s;
}

// ---------------- kernel: fused m-step ----------------
__global__ __launch_bounds__(256) void mstep_kernel(
        const float* __restrict__ act,  const float* __restrict__ pose,
        const float* __restrict__ trans, const float* __restrict__ bias_a,
        const float* __restrict__ bias_b, const float* __restrict__ R,
        const float* __restrict__ denom, float* __restrict__ mu,
        float* __restrict__ sigma, float* __restrict__ logits) {
    __shared__ float ldsT[2][IC * 16 * QS];   // 2 x 16 KB transitions tile
    __shared__ float ldsP[2][IC * 2 * QS];    // 2 x  2 KB pose tile

    const int t = threadIdx.x;
    const int lane = t & 31;
    const int wave = t >> 5;
    const int bp = blockIdx.x >> 3;
    const int obBase = (blockIdx.x & 7) * 16;
    const LaneMap L = make_lane_map(lane, wave, bp, obBase);

    const float den = denom[L.bC * CO + L.oC];
    const float invden = (den == 0.f) ? 0.f : 1.f / den;   // divide_no_nan
    const float* actRow = act + L.bC * CI;
    const float* Rrow   = R + (long)L.bC * CI * CO + L.oC;

    stage_chunk(trans, pose, ldsT[0], ldsP[0], t, bp, obBase, 0);
    wait_async0();
    __syncthreads();

    v8f macc = {}, sacc = {};
    float rsum = 0.f;
    for (int c = 0; c < NCHUNK; ++c) {
        const int buf = c & 1;
        if (c + 1 < NCHUNK)
            stage_chunk(trans, pose, ldsT[buf ^ 1], ldsP[buf ^ 1],
                        t, bp, obBase, (c + 1) * IC);
#pragma unroll
        for (int ii = 0; ii < IC; ++ii) {
            const int i = c * IC + ii;
            const float* lp = ldsP[buf] + ii * (2 * QS);
            const float* lt = ldsT[buf] + ii * (16 * QS);
            v8f v = vote_tile(lp, lt, L);
            const float r = actRow[i] * Rrow[(long)i * CO] * invden;
            rsum += r;
#pragma unroll
            for (int j = 0; j < 8; ++j) {
                const float vj = v[j];
                macc[j] += r * vj;
                sacc[j] += r * vj * vj;
            }
        }
        wait_async0();
        __syncthreads();
    }

    // sigma^2 = sum rV^2 - 2 mu (sum rV) + mu^2 rsum, with mu = sum rV
    float costq = 0.f;
    const int base = ((L.bC * CO + L.oC) << 6) + L.sC;
#pragma unroll
    for (int j = 0; j < 8; ++j) {
        const float m  = macc[j];
        float var = sacc[j] - m * m * (2.f - rsum);
        var = fmaxf(var, 0.f);
        const float sg = sqrtf(var);
        mu[base + (j << 3)]    = m;
        sigma[base + (j << 3)] = sg;
        costq += __logf(sg) * rsum + (float)CI * HALF_PLUS;
    }
    costq += __shfl_xor(costq, 1, 32);
    costq += __shfl_xor(costq, 2, 32);
    costq += __shfl_xor(costq, 4, 32);
    if (L.sC == 0)
        logits[L.bC * CO + L.oC] =
            LAMB * (bias_a[L.oC] - bias_b[L.oC] * rsum - costq);
}

// ---------------- kernel: softmax over CO per batch ----------------
__global__ __launch_bounds__(CO) void softmax_kernel(const float* __restrict__ logits,
                                                     float* __restrict__ act_out) {
    __shared__ float red[CO];
    const int b = blockIdx.x, o = threadIdx.x;
    const float x = logits[b * CO + o];
    red[o] = x; __syncthreads();
    for (int s = CO / 2; s > 0; s >>= 1) {
        if (o < s) red[o] = fmaxf(red[o], red[o + s]);
        __syncthreads();
    }
    const float mx = red[0]; __syncthreads();
    const float e = __expf(x - mx);
    red[o] = e; __syncthreads();
    for (int s = CO / 2; s > 0; s >>= 1) {
        if (o < s) red[o] += red[o + s];
        __syncthreads();
    }
    act_out[b * CO + o] = e / red[0];
}

// ---------------- kernel: fused e-step ----------------
// Writes unnormalized R' = act_out * sum_qs N(V; mu, sigma); the reference's
// global L1 normalization is a uniform scale that cancels in the next m-step.
__global__ __launch_bounds__(256) void estep_kernel(
        const float* __restrict__ pose, const float* __restrict__ trans,
        const float* __restrict__ act_out, const float* __restrict__ mu,
        const float* __restrict__ sigma, float* __restrict__ R) {
    __shared__ float ldsT[2][IC * 16 * QS];
    __shared__ float ldsP[2][IC * 2 * QS];

    const int t = threadIdx.x;
    const int lane = t & 31;
    const int wave = t >> 5;
    const int bp = blockIdx.x >> 3;
    const int obBase = (blockIdx.x & 7) * 16;
    const LaneMap L = make_lane_map(lane, wave, bp, obBase);

    float muv[8], w2[8], cf[8];
    const int base = ((L.bC * CO + L.oC) << 6) + L.sC;
#pragma unroll
    for (int j = 0; j < 8; ++j) {
        const float m  = mu[base + (j << 3)];
        const float sg = sigma[base + (j << 3)];
        muv[j] = m;
        w2[j]  = -0.5f / (sg * sg);
        cf[j]  = 1.f / (sg * SQRT_2PI);
    }
    const float ao = act_out[L.bC * CO + L.oC];
    float* Rw = R + (long)L.bC * CI * CO + L.oC;

    stage_chunk(trans, pose, ldsT[0], ldsP[0], t, bp, obBase, 0);
    wait_async0();
    __syncthreads();

    for (int c = 0; c < NCHUNK; ++c) {
        const int buf = c & 1;
        if (c + 1 < NCHUNK)
            stage_chunk(trans, pose, ldsT[buf ^ 1], ldsP[buf ^ 1],
                        t, bp, obBase, (c + 1) * IC);
#pragma unroll
        for (int ii = 0; ii < IC; ++ii) {
            const int i = c * IC + ii;
            const float* lp = ldsP[buf] + ii * (2 * QS);
            const float* lt = ldsT[buf] + ii * (16 * QS);
            v8f v = vote_tile(lp, lt, L);
            float ps = 0.f;
#pragma unroll
            for (int j = 0; j < 8; ++j) {
                const float d = v[j] - muv[j];
                ps += __expf(d * d * w2[j]) * cf[j];
            }
            ps += __shfl_xor(ps, 1, 32);
            ps += __shfl_xor(ps, 2, 32);
            ps += __shfl_xor(ps, 4, 32);
            if (L.sC == 0) Rw[(long)i * CO] = ao * ps;
        }
        wait_async0();
        __syncthreads();
    }
}

// ---------------- host orchestration ----------------
extern "C" void kernel_launch(void* const* d_in, const int* in_sizes, int n_in,
                              void* d_out, int out_size, void* d_ws, size_t ws_size,
                              hipStream_t stream) {
    const float* act    = (const float*)d_in[0];   // [16,512]
    const float* pose   = (const float*)d_in[1];   // [16,512,8,8]
    const float* trans  = (const float*)d_in[2];   // [512,128,8,8]
    const float* bias_a = (const float*)d_in[3];   // [128]
    const float* bias_b = (const float*)d_in[4];   // [128]

    float* R       = (float*)d_ws;                 // [16,512,128]  4 MB
    float* denom   = R + (long)BATCH * CI * CO;    // [16,128]
    float* mu      = denom + BATCH * CO;           // [16,128,64]
    float* sigma   = mu + BATCH * CO * QS;         // [16,128,64]
    float* logits  = sigma + BATCH * CO * QS;      // [16,128]
    float* act_out = logits + BATCH * CO;          // [16,128]

    initR_kernel<<<(BATCH * CI * CO) / 256, 256, 0, stream>>>(R);
    for (int it = 0; it < 3; ++it) {
        if (it > 0)
            estep_kernel<<<64, 256, 0, stream>>>(pose, trans, act_out, mu, sigma, R);
        denom_kernel<<<BATCH, CO, 0, stream>>>(act, R, denom);
        mstep_kernel<<<64, 256, 0, stream>>>(act, pose, trans, bias_a, bias_b,
                                             R, denom, mu, sigma, logits);
        softmax_kernel<<<BATCH, CO, 0, stream>>>(logits, act_out);
    }
    // outputs: act_out [16,128] then mu [16,128,8,8], concatenated flat
    hipMemcpyAsync(d_out, act_out, (size_t)BATCH * CO * sizeof(float),
                   hipMemcpyDeviceToDevice, stream);
    hipMemcpyAsync((float*)d_out + BATCH * CO, mu,
                   (size_t)BATCH * CO * QS * sizeof(float),
                   hipMemcpyDeviceToDevice, stream);
}